// GraphCNN_47811575939605
// MI455X (gfx1250) — compile-verified
//
#include <hip/hip_runtime.h>

#define N_NODES    50000
#define N_EDGES    800000
#define DIM        128
#define N_LAYERS   2
#define NUM_GRAPHS 512
#define BN_EPS     1e-5f

#define ROWS_PER_WAVE   16
#define WAVES_PER_BLOCK 4
#define N_TILES         (N_NODES / ROWS_PER_WAVE)   // 3125 exact
#define LDS_STRIDE      132                          // 128 + 4 pad -> conflict-free

typedef __attribute__((ext_vector_type(2))) float v2f;
typedef __attribute__((ext_vector_type(8))) float v8f;

// ---------------------------------------------------------------------------
// pooled[dst] += w[e] * h[src]   (one edge per 32 lanes, 4 floats per lane)
// ---------------------------------------------------------------------------
__global__ __launch_bounds__(256)
void gin_scatter(const float* __restrict__ h,
                 const int*   __restrict__ esrc,
                 const int*   __restrict__ edst,
                 const float* __restrict__ ew,
                 float*       __restrict__ pooled) {
    long long tid  = (long long)blockIdx.x * blockDim.x + threadIdx.x;
    int       e    = (int)(tid >> 5);
    int       lane = (int)(tid & 31);
    if (e >= N_EDGES) return;
    float w = ew[e];
    int   s = esrc[e];
    int   d = edst[e];
    const float4 v = *(const float4*)(h + (size_t)s * DIM + lane * 4);
    float* p = pooled + (size_t)d * DIM + lane * 4;
    unsafeAtomicAdd(p + 0, w * v.x);
    unsafeAtomicAdd(p + 1, w * v.y);
    unsafeAtomicAdd(p + 2, w * v.z);
    unsafeAtomicAdd(p + 3, w * v.w);
}

// ---------------------------------------------------------------------------
// Fused MLP: z = relu(pooled @ W1 + b1) @ W2 + b2, plus per-channel BN partial
// sums. One wave per 16-row tile; fp32 WMMA 16x16x4 for exact-precision GEMM.
// ---------------------------------------------------------------------------
__global__ __launch_bounds__(WAVES_PER_BLOCK * 32)
void gin_mlp_wmma(const float* __restrict__ pooled,
                  const float* __restrict__ W1, const float* __restrict__ b1,
                  const float* __restrict__ W2, const float* __restrict__ b2,
                  float* __restrict__ z,
                  float* __restrict__ bn_sum, float* __restrict__ bn_sumsq) {
    __shared__ float tile[WAVES_PER_BLOCK][ROWS_PER_WAVE * LDS_STRIDE];

    const int wave = threadIdx.x >> 5;
    const int lane = threadIdx.x & 31;
    const int half = lane >> 4;   // K/2-half for A,B frags; row-half for C
    const int mn   = lane & 15;   // A: row m   |   B,C: column n

    int  tileIdx = blockIdx.x * WAVES_PER_BLOCK + wave;
    bool valid   = (tileIdx < N_TILES);
    if (!valid) tileIdx = 0;      // clamp (no early return: keep barriers safe)
    const int m0  = tileIdx * ROWS_PER_WAVE;
    float*    lds = tile[wave];

    // --- stage pooled 16x128 tile into LDS (coalesced float4) ---------------
#pragma unroll
    for (int r = 0; r < ROWS_PER_WAVE; ++r) {
        const float4 v = *(const float4*)(pooled + (size_t)(m0 + r) * DIM + lane * 4);
        *(float4*)(lds + r * LDS_STRIDE + lane * 4) = v;
    }
    __syncthreads();

    // --- GEMM1: hid = relu(pooled @ W1 + b1), kept in registers -------------
    v8f hidAcc[8];
#pragma unroll
    for (int nt = 0; nt < 8; ++nt) {
        const int    n  = nt * 16 + mn;
        const float  bv = b1[n];
        v8f acc;
#pragma unroll
        for (int v = 0; v < 8; ++v) acc[v] = bv;
        const float* Wc = W1 + n;  // column n of W1 (row-major [K][N])
#pragma unroll 4
        for (int k = 0; k < DIM; k += 4) {
            v2f a = *(const v2f*)(lds + mn * LDS_STRIDE + k + 2 * half);
            v2f b;
            b.x = Wc[(size_t)(k + 2 * half + 0) * DIM];
            b.y = Wc[(size_t)(k + 2 * half + 1) * DIM];
            acc = __builtin_amdgcn_wmma_f32_16x16x4_f32(
                      false, a, false, b, (short)0, acc, false, false);
        }
#pragma unroll
        for (int v = 0; v < 8; ++v) acc[v] = fmaxf(acc[v], 0.0f);
        hidAcc[nt] = acc;
    }
    __syncthreads();

    // --- store hid to LDS (C-layout -> row-major, padded stride) ------------
#pragma unroll
    for (int nt = 0; nt < 8; ++nt) {
        v8f acc = hidAcc[nt];
#pragma unroll
        for (int v = 0; v < 8; ++v) {
            const int r = half * 8 + v;
            lds[r * LDS_STRIDE + nt * 16 + mn] = acc[v];
        }
    }
    __syncthreads();

    // --- GEMM2: zt = hid @ W2 + b2; emit z + BN partial stats ---------------
#pragma unroll
    for (int nt = 0; nt < 8; ++nt) {
        const int    n  = nt * 16 + mn;
        const float  bv = b2[n];
        v8f acc;
#pragma unroll
        for (int v = 0; v < 8; ++v) acc[v] = bv;
        const float* Wc = W2 + n;
#pragma unroll 4
        for (int k = 0; k < DIM; k += 4) {
            v2f a = *(const v2f*)(lds + mn * LDS_STRIDE + k + 2 * half);
            v2f b;
            b.x = Wc[(size_t)(k + 2 * half + 0) * DIM];
            b.y = Wc[(size_t)(k + 2 * half + 1) * DIM];
            acc = __builtin_amdgcn_wmma_f32_16x16x4_f32(
                      false, a, false, b, (short)0, acc, false, false);
        }
        if (valid) {
            float s = 0.0f, s2 = 0.0f;
#pragma unroll
            for (int v = 0; v < 8; ++v) {
                const int r = half * 8 + v;
                const float zz = acc[v];
                z[(size_t)(m0 + r) * DIM + n] = zz;
                s  += zz;
                s2 += zz * zz;
            }
            unsafeAtomicAdd(&bn_sum[n],   s);   // both half-waves add -> full col sum
            unsafeAtomicAdd(&bn_sumsq[n], s2);
        }
    }
}

// ---------------------------------------------------------------------------
// BN stats finalize: mu, inv_std per channel
// ---------------------------------------------------------------------------
__global__ __launch_bounds__(128)
void bn_finalize(const float* __restrict__ sum, const float* __restrict__ sumsq,
                 float* __restrict__ mu, float* __restrict__ inv) {
    int c = threadIdx.x;
    if (c < DIM) {
        float m   = sum[c] * (1.0f / N_NODES);
        float var = sumsq[c] * (1.0f / N_NODES) - m * m;
        mu[c]  = m;
        inv[c] = rsqrtf(var + BN_EPS);
    }
}

// ---------------------------------------------------------------------------
// h = relu(gamma * (z - mu) * inv + beta), elementwise
// ---------------------------------------------------------------------------
__global__ __launch_bounds__(256)
void bn_apply_relu(const float* __restrict__ z,
                   const float* __restrict__ mu, const float* __restrict__ inv,
                   const float* __restrict__ gamma, const float* __restrict__ beta,
                   float* __restrict__ hout) {
    size_t i = (size_t)blockIdx.x * blockDim.x + threadIdx.x;
    if (i >= (size_t)N_NODES * DIM) return;
    int   c = (int)(i & (DIM - 1));
    float v = gamma[c] * (z[i] - mu[c]) * inv[c] + beta[c];
    hout[i] = fmaxf(v, 0.0f);
}

// ---------------------------------------------------------------------------
// out[g] += pool_w[i] * h[i]   (one node per 32 lanes)
// ---------------------------------------------------------------------------
__global__ __launch_bounds__(256)
void graph_pool(const float* __restrict__ h, const int* __restrict__ gid,
                const float* __restrict__ pw, float* __restrict__ out) {
    long long tid  = (long long)blockIdx.x * blockDim.x + threadIdx.x;
    int       node = (int)(tid >> 5);
    int       lane = (int)(tid & 31);
    if (node >= N_NODES) return;
    float w = pw[node];
    int   g = gid[node];
    const float4 v = *(const float4*)(h + (size_t)node * DIM + lane * 4);
    float* o = out + (size_t)g * DIM + lane * 4;
    unsafeAtomicAdd(o + 0, w * v.x);
    unsafeAtomicAdd(o + 1, w * v.y);
    unsafeAtomicAdd(o + 2, w * v.z);
    unsafeAtomicAdd(o + 3, w * v.w);
}

// ---------------------------------------------------------------------------
extern "C" void kernel_launch(void* const* d_in, const int* in_sizes, int n_in,
                              void* d_out, int out_size, void* d_ws, size_t ws_size,
                              hipStream_t stream) {
    const float* x     = (const float*)d_in[0];
    const int*   esrc  = (const int*)  d_in[1];
    const int*   edst  = (const int*)  d_in[2];
    const float* ew    = (const float*)d_in[3];
    const int*   gid   = (const int*)  d_in[4];
    const float* pw    = (const float*)d_in[5];
    const float* W1    = (const float*)d_in[6];
    const float* b1    = (const float*)d_in[7];
    const float* W2    = (const float*)d_in[8];
    const float* b2    = (const float*)d_in[9];
    const float* gamma = (const float*)d_in[10];
    const float* beta  = (const float*)d_in[11];
    float*       out   = (float*)d_out;

    const size_t ND = (size_t)N_NODES * DIM;   // 6,400,000 floats
    float* ws     = (float*)d_ws;
    float* pooled = ws;                         // [N, D]
    float* zbuf   = ws + ND;                    // [N, D]
    float* h1     = ws + 2 * ND;                // [N, D] (layer-0 output)
    float* stats  = ws + 3 * ND;                // sum[128] sumsq[128] mu[128] inv[128]
    float* bn_sum = stats;
    float* bn_sq  = stats + DIM;
    float* bn_mu  = stats + 2 * DIM;
    float* bn_inv = stats + 3 * DIM;

    const int scatterBlocks = (N_EDGES * 32) / 256;                       // 100000
    const int mlpBlocks     = (N_TILES + WAVES_PER_BLOCK - 1) / WAVES_PER_BLOCK; // 782
    const int ewBlocks      = (int)((ND + 255) / 256);                    // 25000
    const int poolBlocks    = (N_NODES * 32) / 256;                       // 6250

    const float* h_in = x;
    for (int l = 0; l < N_LAYERS; ++l) {
        hipMemsetAsync(pooled, 0, ND * sizeof(float), stream);
        hipMemsetAsync(stats,  0, 2 * DIM * sizeof(float), stream);

        gin_scatter<<<scatterBlocks, 256, 0, stream>>>(h_in, esrc, edst, ew, pooled);

        gin_mlp_wmma<<<mlpBlocks, WAVES_PER_BLOCK * 32, 0, stream>>>(
            pooled,
            W1 + (size_t)l * DIM * DIM, b1 + (size_t)l * DIM,
            W2 + (size_t)l * DIM * DIM, b2 + (size_t)l * DIM,
            zbuf, bn_sum, bn_sq);

        bn_finalize<<<1, 128, 0, stream>>>(bn_sum, bn_sq, bn_mu, bn_inv);

        float* hout = (l == N_LAYERS - 1) ? (out + (size_t)NUM_GRAPHS * DIM) : h1;
        bn_apply_relu<<<ewBlocks, 256, 0, stream>>>(
            zbuf, bn_mu, bn_inv, gamma + (size_t)l * DIM, beta + (size_t)l * DIM, hout);
        h_in = hout;
    }

    // pooled_x = segment_sum(pool_w * h, graph_ids)
    hipMemsetAsync(out, 0, (size_t)NUM_GRAPHS * DIM * sizeof(float), stream);
    graph_pool<<<poolBlocks, 256, 0, stream>>>(h_in, gid, pw, out);
}